// FlowHead_42803644072242
// MI455X (gfx1250) — compile-verified
//
#include <hip/hip_runtime.h>
#include <hip/hip_bf16.h>
#include <math.h>

// ---------------------------------------------------------------------------
// FlowHead: dw3x3(128) -> pw 128->256 -> ReLU -> dw3x3(256) -> pw 256->2
// B=16, H=64, W=128, NCHW fp32. Weights int8 fake-quantized per tensor.
//
// MI455X roofline: ~10 GFLOP vs 23.3 TB/s HBM -> purely memory-bound, so
// (a) keep exact fp32 and use V_WMMA_F32_16X16X4_F32 for the 128->256 GEMM,
// (b) fuse dw1+pw1 and dw2+pw2 so h1/h3 never hit memory:
//     traffic 705 MB -> ~321 MB (and h2 is L2-resident between kernels:
//     128 MB < 192 MB global L2), i.e. ~30us -> ~10us territory.
// ---------------------------------------------------------------------------

typedef __attribute__((ext_vector_type(2))) float v2f;
typedef __attribute__((ext_vector_type(8))) float v8f;

#define B_    16
#define H_    64
#define W_    128
#define HW_   (H_ * W_)     // 8192
#define CIN_  128
#define CHID_ 256

#define TH_   4             // pixel tile rows per block
#define TW_   16            // pixel tile cols per block (16 = WMMA M)
#define H1S_  80            // LDS k-stride for h1 tile: 64 px + 16 pad so the
                            // two half-wave K-phases of an A-fetch use
                            // disjoint 16-bank sets (conflict-free ds loads)
#define HTW_  18            // halo'd tile cols (TW_+2)
#define HTH_  6             // halo'd tile rows (TH_+2)
#define CCH_  64            // h2 channel chunk staged in LDS per iteration

// ---------------------------------------------------------------------------
// Per-tensor int8 symmetric fake-quant: scale = max|w|/127,
// wq = clip(rint(w/scale), -128, 127) * scale.  rintf == round-half-even.
// ---------------------------------------------------------------------------
__global__ void fakequant_kernel(const float* __restrict__ w,
                                 float* __restrict__ wq, int n) {
    __shared__ float smax[256];
    float m = 0.0f;
    for (int i = threadIdx.x; i < n; i += 256) m = fmaxf(m, fabsf(w[i]));
    smax[threadIdx.x] = m;
    __syncthreads();
    for (int s = 128; s > 0; s >>= 1) {
        if (threadIdx.x < s)
            smax[threadIdx.x] = fmaxf(smax[threadIdx.x], smax[threadIdx.x + s]);
        __syncthreads();
    }
    float scale = smax[0] * (1.0f / 127.0f);
    float inv = (scale > 0.0f) ? (1.0f / scale) : 0.0f;
    for (int i = threadIdx.x; i < n; i += 256) {
        float q = rintf(w[i] * inv);
        q = fminf(fmaxf(q, -128.0f), 127.0f);
        wq[i] = q * scale;
    }
}

// ---------------------------------------------------------------------------
// Fused dw3x3(128) -> pointwise 128->256 -> ReLU.
// Block = 4x16 pixel tile, 256 threads (8 waves), grid = 16*16*8 = 2048.
// Stage 1: h1 tile (64 px x 128 ch) into LDS (depthwise, SAME-padded).
// Stage 2: per-wave 16(M=cols)x16(N=co) tiles, K=128 chained f32 WMMA.
//   A layout (32-bit 16x4): lanes 0-15 carry k0,k0+1; lanes 16-31 k0+2,k0+3.
//   B mirrors with lanes=N -> one aligned float2 of w[co][k] per lane.
//   D: VGPR q -> col = q + 8*(lane>=16), co = n0 + (lane&15) -> contiguous
//   row-segment stores (2x global_store_b128 after bias+ReLU).
// ---------------------------------------------------------------------------
__global__ __launch_bounds__(256)
void fused_dw1_pw1_kernel(const float* __restrict__ x,
                          const float* __restrict__ dw1q,
                          const float* __restrict__ db1,
                          const float* __restrict__ pw1q,
                          const float* __restrict__ pb1,
                          float* __restrict__ h2) {
    __shared__ float h1s[CIN_ * H1S_];            // 40 KB, [k][px(padded)]
    int bx = blockIdx.x;
    int tx = bx & 7;                              // W_/TW_ = 8
    int ty = (bx >> 3) & 15;                      // H_/TH_ = 16
    int n  = bx >> 7;                             // image
    int gx0 = tx * TW_, gy0 = ty * TH_;
    int tid = threadIdx.y * 32 + threadIdx.x;

    // ---- Stage 1: depthwise conv into LDS -------------------------------
    {
        int px = tid & 63;                        // position in 4x16 tile
        int r = px >> 4, c = px & 15;
        int gy = gy0 + r, gx = gx0 + c;
        for (int ch = tid >> 6; ch < CIN_; ch += 4) {   // wave-uniform ch
            const float* ip = x + (size_t)(n * CIN_ + ch) * HW_;
            const float* wp = dw1q + ch * 9;      // wave-uniform (scalar)
            float s = db1[ch];
#pragma unroll
            for (int dy = -1; dy <= 1; ++dy) {
                int yy = gy + dy;
                if (yy < 0 || yy >= H_) continue;
#pragma unroll
                for (int dx = -1; dx <= 1; ++dx) {
                    int xx = gx + dx;
                    if (xx < 0 || xx >= W_) continue;
                    s = fmaf(ip[yy * W_ + xx], wp[(dy + 1) * 3 + (dx + 1)], s);
                }
            }
            h1s[ch * H1S_ + px] = s;
        }
    }
    __syncthreads();

    // ---- Stage 2: WMMA GEMM (uniform control flow, EXEC all-1) ----------
    int lane = threadIdx.x;
    int half = lane >> 4;                         // K phase select
    int lr   = lane & 15;
    int wave = threadIdx.y;
    for (int j = 0; j < 8; ++j) {                 // 64 tile-jobs / 8 waves
        int job   = wave * 8 + j;
        int mrow  = job >> 4;                     // tile row 0..3
        int ntile = job & 15;                     // co block 0..15
        const float* Bbase  = pw1q + (size_t)(ntile * 16 + lr) * CIN_;
        const float* Albase = &h1s[mrow * 16 + lr];
        v8f acc = {};
#pragma unroll 4
        for (int k0 = 0; k0 < CIN_; k0 += 4) {
            int kb = k0 + 2 * half;
            v2f a, b;
            a.x = Albase[kb * H1S_];
            a.y = Albase[(kb + 1) * H1S_];
            b = *(const v2f*)(Bbase + kb);        // kb even -> 8B aligned
            acc = __builtin_amdgcn_wmma_f32_16x16x4_f32(
                false, a, false, b, (short)0, acc, false, false);
        }
        int co = ntile * 16 + lr;
        float bv = pb1[co];
        float* Obase = h2 + ((size_t)n * CHID_ + co) * HW_
                          + (gy0 + mrow) * W_ + gx0 + 8 * half;
#pragma unroll
        for (int q = 0; q < 8; ++q) {
            float v = acc[q] + bv;
            Obase[q] = v > 0.0f ? v : 0.0f;       // fused ReLU
        }
    }
}

// ---------------------------------------------------------------------------
// Fused dw3x3(256) -> pointwise 256->2.
// Same 4x16 tiling.  h2 halo tile (6x18 px) staged in LDS in 64-channel
// chunks; each thread owns (pixel, 16-channel slice), computes dw2+db2 per
// channel and folds it into the 2-channel dot product with pw2q (weight
// loads are wave-uniform -> scalar path).  LDS tree reduces the 4 slices.
// h3 never exists in memory.
// ---------------------------------------------------------------------------
__global__ __launch_bounds__(256)
void fused_dw2_pw2_kernel(const float* __restrict__ h2,
                          const float* __restrict__ dw2q,
                          const float* __restrict__ db2,
                          const float* __restrict__ pw2q,
                          const float* __restrict__ pb2,
                          float* __restrict__ out) {
    __shared__ float h2s[CCH_ * (HTH_ * HTW_)];   // 27.6 KB
    __shared__ float partial[8 * 64];             // [co*4+slice][px]
    int bx = blockIdx.x;
    int tx = bx & 7, ty = (bx >> 3) & 15, n = bx >> 7;
    int gx0 = tx * TW_, gy0 = ty * TH_;
    int tid = threadIdx.x;                        // 1D block of 256
    int px = tid & 63, cslot = tid >> 6;          // cslot wave-uniform
    int r = px >> 4, c = px & 15;

    float a0 = 0.0f, a1 = 0.0f;
    for (int cc = 0; cc < CHID_; cc += CCH_) {
        // stage halo'd h2 chunk: 64 ch x 108 px (zero-padded at borders)
        for (int e = tid; e < CCH_ * HTH_ * HTW_; e += 256) {  // 27 iters
            int ch = e / (HTH_ * HTW_);
            int p  = e % (HTH_ * HTW_);
            int hr = p / HTW_, hc = p % HTW_;
            int yy = gy0 + hr - 1, xx = gx0 + hc - 1;
            float v = 0.0f;
            if (yy >= 0 && yy < H_ && xx >= 0 && xx < W_)
                v = h2[(size_t)(n * CHID_ + cc + ch) * HW_ + yy * W_ + xx];
            h2s[e] = v;
        }
        __syncthreads();
        // dw2 (+db2) per channel, folded into the 2-wide pointwise reduce
        for (int t = 0; t < 16; ++t) {
            int cl = cslot * 16 + t;
            int ch = cc + cl;
            const float* wp = dw2q + ch * 9;      // wave-uniform
            const float* hp = &h2s[cl * (HTH_ * HTW_) + r * HTW_ + c];
            float s = db2[ch];
#pragma unroll
            for (int dy = 0; dy < 3; ++dy)
#pragma unroll
                for (int dx = 0; dx < 3; ++dx)
                    s = fmaf(hp[dy * HTW_ + dx], wp[dy * 3 + dx], s);
            a0 = fmaf(s, pw2q[ch], a0);
            a1 = fmaf(s, pw2q[CHID_ + ch], a1);
        }
        __syncthreads();                          // before chunk overwrite
    }
    partial[cslot * 64 + px] = a0;
    partial[(4 + cslot) * 64 + px] = a1;
    __syncthreads();
    if (tid < 128) {
        int co = tid >> 6, p = tid & 63;
        float s = pb2[co];
#pragma unroll
        for (int q = 0; q < 4; ++q) s += partial[(co * 4 + q) * 64 + p];
        int rr = p >> 4, ccx = p & 15;
        out[((size_t)n * 2 + co) * HW_ + (gy0 + rr) * W_ + gx0 + ccx] = s;
    }
}

// ---------------------------------------------------------------------------
// Launch.  Workspace (floats):
//   [0     ) dw1q 1152   [4096 ) pw1q 32768   [36864) dw2q 2304
//   [40960 ) pw2q 512    [65536) h2 33554432 (128 MB)      total ~128.3 MB
// ---------------------------------------------------------------------------
extern "C" void kernel_launch(void* const* d_in, const int* in_sizes, int n_in,
                              void* d_out, int out_size, void* d_ws, size_t ws_size,
                              hipStream_t stream) {
    const float* x   = (const float*)d_in[0];
    const float* dw1 = (const float*)d_in[1];
    const float* db1 = (const float*)d_in[2];
    const float* pw1 = (const float*)d_in[3];
    const float* pb1 = (const float*)d_in[4];
    const float* dw2 = (const float*)d_in[5];
    const float* db2 = (const float*)d_in[6];
    const float* pw2 = (const float*)d_in[7];
    const float* pb2 = (const float*)d_in[8];
    float* out = (float*)d_out;

    float* ws   = (float*)d_ws;
    float* dw1q = ws;
    float* pw1q = ws + 4096;
    float* dw2q = ws + 36864;
    float* pw2q = ws + 40960;
    float* h2   = ws + 65536;

    // Fake-quantize all four weight tensors (tiny; one workgroup each).
    fakequant_kernel<<<1, 256, 0, stream>>>(dw1, dw1q, CIN_ * 9);
    fakequant_kernel<<<1, 256, 0, stream>>>(pw1, pw1q, CHID_ * CIN_);
    fakequant_kernel<<<1, 256, 0, stream>>>(dw2, dw2q, CHID_ * 9);
    fakequant_kernel<<<1, 256, 0, stream>>>(pw2, pw2q, 2 * CHID_);

    const int nblocks = B_ * (H_ / TH_) * (W_ / TW_);   // 2048

    // x -> (dw1 -> pw1 -> ReLU) -> h2   [h1 lives only in LDS]
    dim3 bA(32, 8);
    fused_dw1_pw1_kernel<<<nblocks, bA, 0, stream>>>(x, dw1q, db1, pw1q, pb1, h2);

    // h2 -> (dw2 -> pw2) -> out         [h3 never materialized]
    fused_dw2_pw2_kernel<<<nblocks, 256, 0, stream>>>(h2, dw2q, db2, pw2q, pb2, out);
}